// KGATModel_67654324846924
// MI455X (gfx1250) — compile-verified
//
#include <hip/hip_runtime.h>

#define BATCH           16384
#define EMBED           64
#define NNBR            50
#define ROWS_PER_BLOCK  16
#define THREADS         128     // 4 wave32s
#define XSTRIDE         68      // 64 + 4 floats padding -> conflict-free A-frag loads

typedef __attribute__((ext_vector_type(2))) float v2f;
typedef __attribute__((ext_vector_type(4))) float v4f;
typedef __attribute__((ext_vector_type(8))) float v8f;

__global__ __launch_bounds__(THREADS)
void kgat_fused_kernel(const int*   __restrict__ item_idx,
                       const int*   __restrict__ nbr_idx,
                       const float* __restrict__ item_tab,
                       const float* __restrict__ ent_tab,
                       const float* __restrict__ W,
                       const float* __restrict__ bias,
                       float*       __restrict__ out)
{
    __shared__ float lds_x[ROWS_PER_BLOCK * XSTRIDE];
    __shared__ int   lds_nbr[ROWS_PER_BLOCK * NNBR];
    __shared__ int   lds_item[ROWS_PER_BLOCK];

    const int tid      = threadIdx.x;
    const int lane     = tid & 31;
    const int wave     = tid >> 5;
    const int half     = lane >> 4;        // 0: lanes 0-15, 1: lanes 16-31
    const int l15      = lane & 15;
    const int row_base = blockIdx.x * ROWS_PER_BLOCK;

    // ---- Stage indices into LDS (contiguous, coalesced) ----
    for (int i = tid; i < ROWS_PER_BLOCK * NNBR; i += THREADS)
        lds_nbr[i] = nbr_idx[row_base * NNBR + i];
    if (tid < ROWS_PER_BLOCK)
        lds_item[tid] = item_idx[row_base + tid];
    __syncthreads();

    // ---- Gather + mean + item add (bandwidth-critical phase) ----
    // Half-wave per row: 16 lanes x float4 = one 256B embedding row per request.
    for (int r = 0; r < 2; ++r) {
        const int rel = wave * 4 + r * 2 + half;     // row within tile, 0..15
        const int* nb = &lds_nbr[rel * NNBR];
        v4f acc0 = {0.f, 0.f, 0.f, 0.f};
        v4f acc1 = {0.f, 0.f, 0.f, 0.f};
        #pragma unroll 5
        for (int k = 0; k < NNBR; k += 2) {
            const int i0 = nb[k];
            const int i1 = nb[k + 1];
            v4f e0 = *(const v4f*)(ent_tab + (size_t)i0 * EMBED + l15 * 4);
            v4f e1 = *(const v4f*)(ent_tab + (size_t)i1 * EMBED + l15 * 4);
            acc0 += e0;
            acc1 += e1;
        }
        const int it_i = lds_item[rel];
        v4f itv = *(const v4f*)(item_tab + (size_t)it_i * EMBED + l15 * 4);
        v4f x = (acc0 + acc1) * (1.0f / (float)NNBR) + itv;
        *(v4f*)(&lds_x[rel * XSTRIDE + l15 * 4]) = x;
    }
    __syncthreads();

    // ---- WMMA phase: out = relu(x @ W^T + b), exact f32 via V_WMMA_F32_16X16X4_F32 ----
    // Each wave computes one 16-col N-tile; 16 K-steps of K=4.
    const int koff = half * 2;                 // A/B K sub-offset per ISA layout
    const int n    = wave * 16 + l15;          // output column

    // B-fragment: B[k][n] = W[n][k]  (Wt), per-lane float2 at (n, kk*4+koff)
    v2f wfrag[16];
    #pragma unroll
    for (int kk = 0; kk < 16; ++kk)
        wfrag[kk] = *(const v2f*)(W + n * EMBED + kk * 4 + koff);

    v8f c = {0.f, 0.f, 0.f, 0.f, 0.f, 0.f, 0.f, 0.f};
    #pragma unroll
    for (int kk = 0; kk < 16; ++kk) {
        // A-fragment: lanes 0-15 hold M=lane, K=kk*4+{0,1}; lanes 16-31 K=kk*4+{2,3}
        v2f a = *(const v2f*)(&lds_x[l15 * XSTRIDE + kk * 4 + koff]);
        c = __builtin_amdgcn_wmma_f32_16x16x4_f32(
                /*neg_a=*/false, a,
                /*neg_b=*/false, wfrag[kk],
                /*c_mod=*/(short)0, c,
                /*reuse_a=*/false, /*reuse_b=*/false);
    }

    // ---- Bias + ReLU + store. D layout: VGPR j -> M = j + half*8, N = l15 ----
    const float bv = bias[n];
    const int mrow = row_base + half * 8;
    #pragma unroll
    for (int j = 0; j < 8; ++j) {
        float v = c[j] + bv;
        v = v > 0.0f ? v : 0.0f;
        out[(size_t)(mrow + j) * EMBED + n] = v;
    }
}

extern "C" void kernel_launch(void* const* d_in, const int* in_sizes, int n_in,
                              void* d_out, int out_size, void* d_ws, size_t ws_size,
                              hipStream_t stream)
{
    const int*   item_idx = (const int*)  d_in[0];
    const int*   nbr_idx  = (const int*)  d_in[1];
    const float* item_tab = (const float*)d_in[2];
    const float* ent_tab  = (const float*)d_in[3];
    const float* W        = (const float*)d_in[4];
    const float* bias     = (const float*)d_in[5];
    float* out = (float*)d_out;

    dim3 grid(BATCH / ROWS_PER_BLOCK);   // 1024 blocks, each a 16-row tile
    dim3 block(THREADS);
    kgat_fused_kernel<<<grid, block, 0, stream>>>(item_idx, nbr_idx, item_tab,
                                                  ent_tab, W, bias, out);
}